// Quadrotor_v0_19370302505150
// MI455X (gfx1250) — compile-verified
//
#include <hip/hip_runtime.h>
#include <cstdint>

// ---------------------------------------------------------------------------
// Quadrotor RK4 (4 substeps) on MI455X / gfx1250.
//  - 2 rows per thread packed into float2 ext-vectors (packed/dual FP32 VALU)
//  - block tile staged in & drained out of LDS with CDNA5 async b128 copies
//    (global_load_async_to_lds_b128 / global_store_async_from_lds_b128)
//  - s_wait_asynccnt + workgroup barriers; ds_load/ds_store b128 in between
// ---------------------------------------------------------------------------

typedef __attribute__((ext_vector_type(2))) float v2f;
typedef __attribute__((ext_vector_type(4))) int   v4i;

typedef __attribute__((address_space(1))) v4i* gbl_v4i_p;  // global v4i*
typedef __attribute__((address_space(3))) v4i* lds_v4i_p;  // LDS v4i*

__device__ __forceinline__ void async_copy_b128(void* lds, const void* g) {
#if __has_builtin(__builtin_amdgcn_global_load_async_to_lds_b128)
  __builtin_amdgcn_global_load_async_to_lds_b128(
      (gbl_v4i_p)(uintptr_t)g,
      (lds_v4i_p)(uint32_t)(uintptr_t)lds,
      0 /*imm offset*/, 0 /*cpol*/);
#else
  asm volatile("global_load_async_to_lds_b128 %0, %1, off"
               :: "v"((uint32_t)(uintptr_t)lds),
                  "v"((uint64_t)(uintptr_t)g)
               : "memory");
#endif
}

__device__ __forceinline__ void async_store_b128(void* g, const void* lds) {
#if __has_builtin(__builtin_amdgcn_global_store_async_from_lds_b128)
  __builtin_amdgcn_global_store_async_from_lds_b128(
      (gbl_v4i_p)(uintptr_t)g,
      (lds_v4i_p)(uint32_t)(uintptr_t)lds,
      0 /*imm offset*/, 0 /*cpol*/);
#else
  asm volatile("global_store_async_from_lds_b128 %0, %1, off"
               :: "v"((uint64_t)(uintptr_t)g),
                  "v"((uint32_t)(uintptr_t)lds)
               : "memory");
#endif
}

__device__ __forceinline__ void wait_async_zero() {
#if __has_builtin(__builtin_amdgcn_s_wait_asynccnt)
  __builtin_amdgcn_s_wait_asynccnt(0);
#else
  asm volatile("s_wait_asynccnt 0" ::: "memory");
#endif
}

// ---------------------------------------------------------------------------
// Math (templated so the same code serves the packed-pair and scalar kernels)
// ---------------------------------------------------------------------------

template <typename T> __device__ __forceinline__ T rsq(T x);
template <> __device__ __forceinline__ float rsq<float>(float x) {
  return __builtin_amdgcn_rsqf(x);  // v_rsq_f32
}
template <> __device__ __forceinline__ v2f rsq<v2f>(v2f x) {
  v2f r;
  r.x = __builtin_amdgcn_rsqf(x.x);
  r.y = __builtin_amdgcn_rsqf(x.y);
  return r;
}

template <typename T>
__device__ __forceinline__ void quad_deriv(const T s[10], T th, T wx, T wy, T wz,
                                           T d[10]) {
  T qw = s[3], qx = s[4], qy = s[5], qz = s[6];
  T inv = rsq<T>(qw * qw + qx * qx + qy * qy + qz * qz);
  qw *= inv; qx *= inv; qy *= inv; qz *= inv;
  d[0] = s[7];
  d[1] = s[8];
  d[2] = s[9];
  d[3] = 0.5f * (-wx * qx - wy * qy - wz * qz);
  d[4] = 0.5f * ( wx * qw + wz * qy - wy * qz);
  d[5] = 0.5f * ( wy * qw - wz * qx + wx * qz);
  d[6] = 0.5f * ( wz * qw + wy * qx - wx * qy);
  d[7] = 2.0f * (qw * qy + qx * qz) * th;
  d[8] = 2.0f * (qy * qz - qw * qx) * th;
  // Unit quaternion: qw^2 - qx^2 - qy^2 + qz^2 == 2*(qw^2 + qz^2) - 1
  d[9] = (2.0f * (qw * qw + qz * qz) - 1.0f) * th - 9.81f;
}

template <typename T>
__device__ __forceinline__ void quad_rk4(T X[10], T th, T wx, T wy, T wz) {
  const float DT = 0.02f / 4.0f;  // 0.005
#pragma unroll
  for (int it = 0; it < 4; ++it) {
    T k[10], acc[10], tmp[10];
    quad_deriv(X, th, wx, wy, wz, k);          // f1
#pragma unroll
    for (int i = 0; i < 10; ++i) { acc[i] = k[i]; tmp[i] = X[i] + (0.5f * DT) * k[i]; }
    quad_deriv(tmp, th, wx, wy, wz, k);        // f2
#pragma unroll
    for (int i = 0; i < 10; ++i) { acc[i] += 2.0f * k[i]; tmp[i] = X[i] + (0.5f * DT) * k[i]; }
    quad_deriv(tmp, th, wx, wy, wz, k);        // f3
#pragma unroll
    for (int i = 0; i < 10; ++i) { acc[i] += 2.0f * k[i]; tmp[i] = X[i] + DT * k[i]; }
    quad_deriv(tmp, th, wx, wy, wz, k);        // f4
#pragma unroll
    for (int i = 0; i < 10; ++i) { X[i] += (DT / 6.0f) * (acc[i] + k[i]); }
  }
}

// ---------------------------------------------------------------------------
// Main tiled kernel: 256 threads, 512 rows per block, 2 rows per thread.
// State tile: 512*40B = 1280 float4 ; action tile: 512*16B = 512 float4.
// Thread t reads AND writes only shS[5t..5t+4], so no barrier is needed
// between the row reads and the result write-back; one barrier before the
// coalesced async drain.
// ---------------------------------------------------------------------------

__global__ __launch_bounds__(256) void quad_rk4_tiled(
    const float* __restrict__ X, const float* __restrict__ A,
    float* __restrict__ out) {
  __shared__ float4 shS[1280];
  __shared__ float4 shA[512];

  const int t = threadIdx.x;
  const long long blk = blockIdx.x;

  const float4* X4 = (const float4*)X;
  const float4* A4 = (const float4*)A;

  // Stage the block tile with lane-consecutive (fully coalesced) async copies.
#pragma unroll
  for (int k = 0; k < 5; ++k)
    async_copy_b128(&shS[k * 256 + t], &X4[blk * 1280 + k * 256 + t]);
#pragma unroll
  for (int k = 0; k < 2; ++k)
    async_copy_b128(&shA[k * 256 + t], &A4[blk * 512 + k * 256 + t]);

  wait_async_zero();   // my wave's asyncs landed in LDS
  __syncthreads();     // everyone's asyncs landed

  // This thread's two rows: 20 consecutive floats = 5 float4 (16B aligned).
  float4 s0 = shS[5 * t + 0];
  float4 s1 = shS[5 * t + 1];
  float4 s2 = shS[5 * t + 2];
  float4 s3 = shS[5 * t + 3];
  float4 s4 = shS[5 * t + 4];

  // Pack component c of (row0,row1) into one v2f lane-pair.
  v2f Xs[10];
  Xs[0] = (v2f){s0.x, s2.z};
  Xs[1] = (v2f){s0.y, s2.w};
  Xs[2] = (v2f){s0.z, s3.x};
  Xs[3] = (v2f){s0.w, s3.y};
  Xs[4] = (v2f){s1.x, s3.z};
  Xs[5] = (v2f){s1.y, s3.w};
  Xs[6] = (v2f){s1.z, s4.x};
  Xs[7] = (v2f){s1.w, s4.y};
  Xs[8] = (v2f){s2.x, s4.z};
  Xs[9] = (v2f){s2.y, s4.w};

  float4 a0 = shA[2 * t + 0];
  float4 a1 = shA[2 * t + 1];
  v2f th = (v2f){a0.x, a1.x};
  v2f wx = (v2f){a0.y, a1.y};
  v2f wy = (v2f){a0.z, a1.z};
  v2f wz = (v2f){a0.w, a1.w};

  quad_rk4<v2f>(Xs, th, wx, wy, wz);

  // Write results back into this thread's own tile slots (no hazard).
  shS[5 * t + 0] = make_float4(Xs[0].x, Xs[1].x, Xs[2].x, Xs[3].x);
  shS[5 * t + 1] = make_float4(Xs[4].x, Xs[5].x, Xs[6].x, Xs[7].x);
  shS[5 * t + 2] = make_float4(Xs[8].x, Xs[9].x, Xs[0].y, Xs[1].y);
  shS[5 * t + 3] = make_float4(Xs[2].y, Xs[3].y, Xs[4].y, Xs[5].y);
  shS[5 * t + 4] = make_float4(Xs[6].y, Xs[7].y, Xs[8].y, Xs[9].y);

  __syncthreads();     // whole output tile resident in LDS

  // Drain the tile with lane-consecutive (full cache line) async stores.
  float4* O4 = (float4*)out;
#pragma unroll
  for (int k = 0; k < 5; ++k)
    async_store_b128(&O4[blk * 1280 + k * 256 + t], &shS[k * 256 + t]);
  // S_ENDPGM performs an implicit wait-idle, covering outstanding ASYNCcnt.
}

// Fallback: one row per thread, any B.
__global__ __launch_bounds__(256) void quad_rk4_scalar(
    const float* __restrict__ X, const float* __restrict__ A,
    float* __restrict__ out, long long B) {
  long long r = (long long)blockIdx.x * blockDim.x + threadIdx.x;
  if (r >= B) return;
  float s[10];
#pragma unroll
  for (int i = 0; i < 10; ++i) s[i] = X[r * 10 + i];
  float th = A[r * 4 + 0], wx = A[r * 4 + 1], wy = A[r * 4 + 2], wz = A[r * 4 + 3];
  quad_rk4<float>(s, th, wx, wy, wz);
#pragma unroll
  for (int i = 0; i < 10; ++i) out[r * 10 + i] = s[i];
}

extern "C" void kernel_launch(void* const* d_in, const int* in_sizes, int n_in,
                              void* d_out, int out_size, void* d_ws, size_t ws_size,
                              hipStream_t stream) {
  (void)n_in; (void)out_size; (void)d_ws; (void)ws_size;
  const float* X = (const float*)d_in[0];
  const float* A = (const float*)d_in[1];
  float* out = (float*)d_out;
  const long long B = (long long)in_sizes[0] / 10;

  if ((B % 512) == 0) {
    const int grid = (int)(B / 512);  // B = 4,194,304 -> 8192 blocks
    quad_rk4_tiled<<<grid, 256, 0, stream>>>(X, A, out);
  } else {
    const int grid = (int)((B + 255) / 256);
    quad_rk4_scalar<<<grid, 256, 0, stream>>>(X, A, out, B);
  }
}